// Net_74345883893868
// MI455X (gfx1250) — compile-verified
//
#include <hip/hip_runtime.h>
#include <math.h>

typedef __attribute__((ext_vector_type(2))) float v2f;
typedef __attribute__((ext_vector_type(8))) float v8f;

#define NBATCH 8192
#define SS 30
#define DD 14
#define UPM 28
#define NTHR 128

// LDS layout (float offsets). All 16x16-tile operands use padded 32-row/col
// strides so WMMA K-padding reads hit zero-initialized memory.
enum {
  O_UL  = 0,      // 32x32  ul
  O_UR  = 1024,   // 32x32  ur
  O_C   = 2048,   // 32x32  c = silu(conv)
  O_H   = 3072,   // 32x32  H = P@V, then "pre", then flat tanh(y2) [420]
  O_Q   = 4096,   // 32x32  q, later y1 rows [t*32+d]
  O_K   = 5120,   // 32x32  k, later xn2 rows
  O_V   = 6144,   // 32x32  v, later sLSTM gx rows [t*32+g]
  O_P   = 7168,   // 32x32  scores A, then P, later sLSTM hs [t*32+{0,1}]
  O_X   = 8192,   // 32x16  xn (LN of x), zero padded
  O_WUP = 8704,   // 16x64  [Wup_l | 4 pad | Wup_r | 4 pad]
  O_WQ  = 9728,   // 32x32  Wq padded
  O_WK  = 10752,  // 32x32  Wk padded
  O_WV  = 11776,  // 32x32  Wv padded
  O_GI  = 12800,  // 32     input-gate preacts i_t
  O_GF  = 12832,  // 32     forget-gate preacts f_t
  O_GG  = 12864,  // 32     g_s = i_s - F_s
  O_GM  = 12896,  // 32     G_t = max(0, running max g)
  O_DEN = 12928,  // 32     denom per row
  O_RED = 12960,  // 32     head reduction
  SM_TOT= 12992   // 51,968 bytes
};

struct Args { const float* p[39]; float* out; };

__device__ __forceinline__ float sigm(float x) { return 1.0f / (1.0f + __expf(-x)); }
__device__ __forceinline__ float silu(float x) { return x * sigm(x); }
__device__ __forceinline__ float gelu_t(float x) {
  float x3 = x * x * x;
  return 0.5f * x * (1.0f + tanhf(0.7978845608028654f * (x + 0.044715f * x3)));
}

// One 16x16 fp32 WMMA tile: acc += A[rowOff:+16, 0:4*ksteps] * B[0:4*ksteps, colOff:+16]
// A operand layout (ISA 16x4 f32): lane m=l&15, low half-wave holds K=4k+0/1, high K=4k+2/3.
// B operand mirrored; trB reads B[k][n] = Bs[n][k] (transposed source, e.g. K^T from k-matrix).
__device__ __forceinline__ void wmma_tile(const float* __restrict__ As, int lda, int rowOff,
                                          const float* __restrict__ Bs, int ldb, int colOff,
                                          int ksteps, bool trB, v8f& acc, int lane) {
  const int m  = lane & 15;
  const int hi = lane >> 4;
  for (int kk = 0; kk < ksteps; ++kk) {
    const int k0 = kk * 4 + 2 * hi;
    v2f a, b;
    a.x = As[(rowOff + m) * lda + k0];
    a.y = As[(rowOff + m) * lda + k0 + 1];
    if (!trB) {
      b.x = Bs[(k0    ) * ldb + colOff + m];
      b.y = Bs[(k0 + 1) * ldb + colOff + m];
    } else {
      b.x = Bs[(colOff + m) * ldb + k0];
      b.y = Bs[(colOff + m) * ldb + k0 + 1];
    }
    acc = __builtin_amdgcn_wmma_f32_16x16x4_f32(false, a, false, b, (short)0, acc,
                                                false, false);
  }
}

// C/D layout: VGPR r -> rows r (lanes 0-15) and r+8 (lanes 16-31), col = lane&15.
__device__ __forceinline__ void store_tile(float* __restrict__ Os, int ldo, int rowOff,
                                           int colOff, const v8f& acc,
                                           const float* __restrict__ bias, int ncols,
                                           float scale, int lane) {
  const int n  = lane & 15;
  const int hi = lane >> 4;
#pragma unroll
  for (int r = 0; r < 8; ++r) {
    const int row = rowOff + r + 8 * hi;
    const int col = colOff + n;
    float v = 0.0f;
    if (col < ncols) {
      v = acc[r];
      if (bias) v += bias[col];
      v *= scale;
    }
    Os[row * ldo + col] = v;
  }
}

__global__ void __launch_bounds__(NTHR) xlstm_fused(Args A) {
  __shared__ float sm[SM_TOT];
  const int tid  = threadIdx.x;
  const int lane = tid & 31;
  const int wv   = tid >> 5;
  const int b    = blockIdx.x;

  const float* x     = A.p[0];
  const float* mlng  = A.p[1];  const float* mlnb  = A.p[2];
  const float* wupl  = A.p[3];  const float* bupl  = A.p[4];
  const float* wupr  = A.p[5];  const float* bupr  = A.p[6];
  const float* convw = A.p[7];  const float* convb = A.p[8];
  const float* wq    = A.p[9];  const float* bq    = A.p[10];
  const float* wk    = A.p[11]; const float* bk    = A.p[12];
  const float* wvv   = A.p[13]; const float* bv    = A.p[14];
  const float* wif   = A.p[15]; const float* bif   = A.p[16];
  const float* gng   = A.p[17]; const float* gnb   = A.p[18];
  const float* skip  = A.p[19];
  const float* wdn   = A.p[20]; const float* bdn   = A.p[21];
  const float* slng  = A.p[22]; const float* slnb  = A.p[23];
  const float* sW    = A.p[24]; const float* sR    = A.p[25]; const float* sb = A.p[26];
  const float* sgng  = A.p[27]; const float* sgnb  = A.p[28];
  const float* sup1  = A.p[29]; const float* sbu1  = A.p[30];
  const float* sup2  = A.p[31]; const float* sbu2  = A.p[32];
  const float* sdn   = A.p[33]; const float* sbd   = A.p[34];
  const float* fc1w  = A.p[35]; const float* fc1b  = A.p[36];
  const float* fc3w  = A.p[37]; const float* fc3b  = A.p[38];

  // ---- Phase 0: zero LDS (exact zero padding for WMMA K-dims), stage weights ----
  for (int i = tid; i < SM_TOT; i += NTHR) sm[i] = 0.0f;
  __syncthreads();
  for (int idx = tid; idx < 1024; idx += NTHR) {      // [Wup_l|Wup_r] 16x64 padded
    int r = idx >> 6, cg = idx & 63;
    float v = 0.0f;
    if (r < DD) {
      if (cg < UPM)                    v = wupl[r * UPM + cg];
      else if (cg >= 32 && cg < 32+UPM) v = wupr[r * UPM + (cg - 32)];
    }
    sm[O_WUP + idx] = v;
  }
  for (int idx = tid; idx < 1024; idx += NTHR) {      // Wq/Wk/Wv 32x32 padded
    int r = idx >> 5, c = idx & 31;
    float vq = 0.0f, vk = 0.0f, vv = 0.0f;
    if (r < UPM && c < UPM) {
      vq = wq[r * UPM + c]; vk = wk[r * UPM + c]; vv = wvv[r * UPM + c];
    }
    sm[O_WQ + idx] = vq; sm[O_WK + idx] = vk; sm[O_WV + idx] = vv;
  }
  // warm the head weights (used last): global_prefetch_b8
  for (int i = tid * 64; i < 420 * 32 * 4; i += NTHR * 64)
    __builtin_prefetch((const char*)fc1w + i, 0, 1);
  __syncthreads();

  // ---- Phase 1: xn = LN(x) -> sX (30x14 in 32x16, pads stay 0) ----
  if (tid < SS) {
    const float* xr = x + ((size_t)b * SS + tid) * DD;
    float v[DD]; float mu = 0.0f;
    for (int d = 0; d < DD; ++d) { v[d] = xr[d]; mu += v[d]; }
    mu *= (1.0f / DD);
    float va = 0.0f;
    for (int d = 0; d < DD; ++d) { float t = v[d] - mu; va += t * t; }
    va *= (1.0f / DD);
    float rs = rsqrtf(va + 1e-5f);
    for (int d = 0; d < DD; ++d)
      sm[O_X + tid * 16 + d] = (v[d] - mu) * rs * mlng[d] + mlnb[d];
  }
  __syncthreads();

  // ---- Phase 2 (WMMA): [ul|ur] = xn @ [Wup_l|Wup_r], 8 tiles / 4 waves ----
  for (int i = 0; i < 2; ++i) {
    int t = wv + 4 * i;
    int rt = t >> 2, ct = t & 3;
    v8f acc = {};
    wmma_tile(sm + O_X, 16, rt * 16, sm + O_WUP, 64, ct * 16, 4, false, acc, lane);
    int n = lane & 15, hi = lane >> 4;
#pragma unroll
    for (int r = 0; r < 8; ++r) {
      int row = rt * 16 + r + 8 * hi;
      int cg  = ct * 16 + n;
      if (cg < UPM)                     sm[O_UL + row * 32 + cg]        = acc[r] + bupl[cg];
      else if (cg >= 32 && cg < 32+UPM) sm[O_UR + row * 32 + (cg - 32)] = acc[r] + bupr[cg - 32];
    }
  }
  __syncthreads();

  // ---- Phase 3: causal depthwise conv (K=4) + SiLU -> c ----
  for (int idx = tid; idx < SS * UPM; idx += NTHR) {
    int t = idx / UPM, j = idx - t * UPM;
    float acc = convb[j];
#pragma unroll
    for (int k = 0; k < 4; ++k) {
      int tt = t - 3 + k;
      if (tt >= 0) acc += sm[O_UL + tt * 32 + j] * convw[k * UPM + j];
    }
    sm[O_C + t * 32 + j] = silu(acc);
  }
  __syncthreads();

  // ---- Phase 4 (WMMA): q = c@Wq, k = (c@Wk)/sqrt(28), v = ul@Wv ; + gif gates ----
  for (int i = 0; i < 3; ++i) {
    int id = wv + 4 * i;
    int mat = id >> 2, sub = id & 3;
    int rt = sub >> 1, ct = sub & 1;
    const float* Asrc = (mat == 2) ? (sm + O_UL) : (sm + O_C);
    const float* Bsrc = (mat == 0) ? (sm + O_WQ) : (mat == 1 ? sm + O_WK : sm + O_WV);
    float*       Dst  = (mat == 0) ? (sm + O_Q)  : (mat == 1 ? sm + O_K  : sm + O_V);
    const float* bias = (mat == 0) ? bq : (mat == 1 ? bk : bv);
    float scale = (mat == 1) ? 0.1889822365046136f : 1.0f;   // 1/sqrt(UP_M)
    v8f acc = {};
    wmma_tile(Asrc, 32, rt * 16, Bsrc, 32, ct * 16, 8, false, acc, lane);
    store_tile(Dst, 32, rt * 16, ct * 16, acc, bias, UPM, scale, lane);
  }
  if (tid < 2 * SS) {   // gif = c @ Wif + bif
    int t = tid >> 1, g = tid & 1;
    float acc = bif[g];
    for (int j = 0; j < UPM; ++j) acc += sm[O_C + t * 32 + j] * wif[j * 2 + g];
    sm[(g == 0 ? O_GI : O_GF) + t] = acc;
  }
  __syncthreads();

  // ---- Phase 5: gate prefix scan (serial, tiny): g_s = i_s - F_s, G_t = max(0, cummax g) ----
  if (tid == 0) {
    float F = 0.0f, gmax = -3.0e38f;
    for (int t = 0; t < SS; ++t) {
      F += sm[O_GF + t];
      float g = sm[O_GI + t] - F;
      gmax = fmaxf(gmax, g);
      sm[O_GG + t] = g;
      sm[O_GM + t] = fmaxf(0.0f, gmax);
    }
  }
  __syncthreads();

  // ---- Phase 6 (WMMA): scores A = Q K^T (B transposed from k-matrix), 4 tiles ----
  {
    int rt = wv >> 1, ct = wv & 1;
    v8f acc = {};
    wmma_tile(sm + O_Q, 32, rt * 16, sm + O_K, 32, ct * 16, 8, true, acc, lane);
    store_tile(sm + O_P, 32, rt * 16, ct * 16, acc, nullptr, 32, 1.0f, lane);
  }
  __syncthreads();

  // ---- Phase 7: P = causal-mask(W ∘ A), then row sums -> denom ----
  for (int idx = tid; idx < SS * 32; idx += NTHR) {
    int t = idx >> 5, s = idx & 31;
    float w = 0.0f;
    if (s <= t && s < SS)
      w = sm[O_P + idx] * __expf(sm[O_GG + s] - sm[O_GM + t]);
    sm[O_P + idx] = w;
  }
  __syncthreads();
  if (tid < SS) {
    float sum = 0.0f;
    for (int s = 0; s < SS; ++s) sum += sm[O_P + tid * 32 + s];
    sm[O_DEN + tid] = fmaxf(fabsf(sum), 1.0f);
  }
  __syncthreads();

  // ---- Phase 8 (WMMA): H = P @ V, 4 tiles ----
  {
    int rt = wv >> 1, ct = wv & 1;
    v8f acc = {};
    wmma_tile(sm + O_P, 32, rt * 16, sm + O_V, 32, ct * 16, 8, false, acc, lane);
    store_tile(sm + O_H, 32, rt * 16, ct * 16, acc, nullptr, 32, 1.0f, lane);
  }
  __syncthreads();

  // ---- Phase 9: groupnorm(h/den) + skip*c, gated by silu(ur) -> pre (in sH) ----
  if (tid < SS) {
    int t = tid;
    float inv = 1.0f / sm[O_DEN + t];
    float hv[UPM]; float mu = 0.0f;
    for (int j = 0; j < UPM; ++j) { hv[j] = sm[O_H + t * 32 + j] * inv; mu += hv[j]; }
    mu *= (1.0f / UPM);
    float va = 0.0f;
    for (int j = 0; j < UPM; ++j) { float d = hv[j] - mu; va += d * d; }
    va *= (1.0f / UPM);
    float rs = rsqrtf(va + 1e-5f);
    for (int j = 0; j < UPM; ++j) {
      float hn = (hv[j] - mu) * rs * gng[j] + gnb[j];
      float h2 = hn + skip[j] * sm[O_C + t * 32 + j];
      sm[O_H + t * 32 + j] = h2 * silu(sm[O_UR + t * 32 + j]);
    }
  }
  __syncthreads();
  // y1 = pre @ Wdown + bdown + x   (30x14, VALU)
  for (int idx = tid; idx < SS * DD; idx += NTHR) {
    int t = idx / DD, d = idx - t * DD;
    float acc = bdn[d];
    for (int j = 0; j < UPM; ++j) acc += sm[O_H + t * 32 + j] * wdn[j * DD + d];
    acc += x[((size_t)b * SS + t) * DD + d];
    sm[O_Q + t * 32 + d] = acc;                       // y1 rows
  }
  __syncthreads();

  // ---- Phase 10: sLSTM block (H=2, tiny) ----
  if (tid < SS) {                                     // xn2 = LN(y1)
    int t = tid; float v[DD]; float mu = 0.0f;
    for (int d = 0; d < DD; ++d) { v[d] = sm[O_Q + t * 32 + d]; mu += v[d]; }
    mu *= (1.0f / DD);
    float va = 0.0f;
    for (int d = 0; d < DD; ++d) { float q = v[d] - mu; va += q * q; }
    va *= (1.0f / DD);
    float rs = rsqrtf(va + 1e-5f);
    for (int d = 0; d < DD; ++d)
      sm[O_K + t * 32 + d] = (v[d] - mu) * rs * slng[d] + slnb[d];
  }
  __syncthreads();
  for (int idx = tid; idx < SS * 8; idx += NTHR) {    // gx = xn2 @ sW + sb
    int t = idx >> 3, g = idx & 7;
    float acc = sb[g];
    for (int d = 0; d < DD; ++d) acc += sm[O_K + t * 32 + d] * sW[d * 8 + g];
    sm[O_V + t * 32 + g] = acc;
  }
  __syncthreads();
  if (tid == 0) {                                     // serial recurrence, H=2
    float h[2] = {0, 0}, cs[2] = {0, 0}, nn[2] = {0, 0}, mm[2] = {0, 0};
    for (int t = 0; t < SS; ++t) {
      float g[8];
      for (int j = 0; j < 8; ++j)
        g[j] = sm[O_V + t * 32 + j] + h[0] * sR[j] + h[1] * sR[8 + j];
      for (int j = 0; j < 2; ++j) {
        float z  = tanhf(g[j]);
        float it = g[2 + j], ft = g[4 + j];
        float o  = sigm(g[6 + j]);
        float mn = fmaxf(ft + mm[j], it);
        float ie = __expf(it - mn);
        float fe = __expf(ft + mm[j] - mn);
        cs[j] = fe * cs[j] + ie * z;
        nn[j] = fe * nn[j] + ie;
        mm[j] = mn;
        h[j]  = o * cs[j] / nn[j];
        sm[O_P + t * 32 + j] = h[j];
      }
    }
  }
  __syncthreads();
  if (tid < SS) {                                     // LN(H=2), up/gate, down, tanh(y2)
    int t = tid;
    float h0 = sm[O_P + t * 32 + 0], h1 = sm[O_P + t * 32 + 1];
    float mu = 0.5f * (h0 + h1);
    float d0 = h0 - mu, d1 = h1 - mu;
    float rs = rsqrtf(0.5f * (d0 * d0 + d1 * d1) + 1e-5f);
    float hn0 = d0 * rs * sgng[0] + sgnb[0];
    float hn1 = d1 * rs * sgng[1] + sgnb[1];
    float u[3];
    for (int k = 0; k < 3; ++k) {
      float a1 = hn0 * sup1[k] + hn1 * sup1[3 + k] + sbu1[k];
      float a2 = hn0 * sup2[k] + hn1 * sup2[3 + k] + sbu2[k];
      u[k] = gelu_t(a1) * a2;
    }
    for (int d = 0; d < DD; ++d) {
      float y2 = sbd[d] + u[0] * sdn[d] + u[1] * sdn[DD + d] + u[2] * sdn[2 * DD + d]
               + sm[O_Q + t * 32 + d];
      sm[O_H + t * DD + d] = tanhf(y2);               // flat [420]
    }
  }
  __syncthreads();

  // ---- Phase 11: head  relu(flat @ fc1 + b) @ fc3 + b ----
  if (tid < 32) {
    float acc = fc1b[tid];
    for (int i = 0; i < SS * DD; ++i) acc += sm[O_H + i] * fc1w[i * 32 + tid];
    sm[O_RED + tid] = fmaxf(acc, 0.0f) * fc3w[tid];
  }
  __syncthreads();
  if (tid == 0) {
    float s = fc3b[0];
    for (int j = 0; j < 32; ++j) s += sm[O_RED + j];
    A.out[b] = s;
  }
}

extern "C" void kernel_launch(void* const* d_in, const int* in_sizes, int n_in,
                              void* d_out, int out_size, void* d_ws, size_t ws_size,
                              hipStream_t stream) {
  (void)in_sizes; (void)n_in; (void)out_size; (void)d_ws; (void)ws_size;
  Args a;
  for (int i = 0; i < 39; ++i) a.p[i] = (const float*)d_in[i];
  a.out = (float*)d_out;
  xlstm_fused<<<dim3(NBATCH), dim3(NTHR), 0, stream>>>(a);
}